// AFF_Deform_39230231281672
// MI455X (gfx1250) — compile-verified
//
#include <hip/hip_runtime.h>
#include <hip/hip_bf16.h>

typedef float v2f __attribute__((ext_vector_type(2)));
typedef float v8f __attribute__((ext_vector_type(8)));

#define BB_  8
#define HH   128
#define WW   128
#define CIN  192
#define CO   64
#define LDK1 196   // padded K-stride, stage-2 LDS tiles (196 % 64 == 4 -> conflict-free)
#define LDK2 580   // padded K-stride, stage-4 sample tile (580 % 64 == 4)
#define LDB4 148   // padded K-stride, stage-4 weight chunk (148 % 64 == 20 -> conflict-free)

// ---------------------------------------------------------------------------
// Kernel 1: off1 = 1x1 conv (implicit concat of x1,x2,x4; 192 -> 2)
// ---------------------------------------------------------------------------
__global__ void __launch_bounds__(256)
k_off1(const float* __restrict__ x1, const float* __restrict__ x2,
       const float* __restrict__ x4, const float* __restrict__ wt,
       const float* __restrict__ bias, float* __restrict__ off1)
{
    const int idx = blockIdx.x * blockDim.x + threadIdx.x;   // 131072 pixels
    const int w = idx & (WW - 1);
    const int h = (idx >> 7) & (HH - 1);
    const int b = idx >> 14;
    const size_t plane = (size_t)HH * WW;
    const size_t pix = (size_t)h * WW + w;

    float a0 = 0.f, a1 = 0.f;
    const float* xs[3] = { x1, x2, x4 };
#pragma unroll
    for (int g3 = 0; g3 < 3; ++g3) {
        const float* xp = xs[g3] + ((size_t)b * 64) * plane + pix;
        for (int c = 0; c < 64; ++c) {
            const float v = xp[(size_t)c * plane];
            a0 = fmaf(v, wt[g3 * 64 + c], a0);            // uniform -> s_load
            a1 = fmaf(v, wt[CIN + g3 * 64 + c], a1);
        }
    }
    off1[(((size_t)b * 2 + 0) * HH + h) * WW + w] = a0 + bias[0];
    off1[(((size_t)b * 2 + 1) * HH + h) * WW + w] = a1 + bias[1];
}

// ---------------------------------------------------------------------------
// Kernel 2: deform-conv k=1 (192 -> 64 GEMM via WMMA f32 16x16x4) + BN + ReLU
// Block: 128 threads (4 waves), tile = 64 pixels of one (b,h) row.
// Both A (samples) and B (w1) staged in LDS; inner loop is pure DS + WMMA.
// ---------------------------------------------------------------------------
__global__ void __launch_bounds__(128)
k_deform1(const float* __restrict__ x1, const float* __restrict__ x2,
          const float* __restrict__ x4, const float* __restrict__ off1,
          const float* __restrict__ w1, const float* __restrict__ b1,
          const float* __restrict__ bng, const float* __restrict__ bnb,
          const float* __restrict__ bnm, const float* __restrict__ bnv,
          float* __restrict__ y1)
{
    extern __shared__ char smem[];
    float* S  = (float*)smem;                                   // [64][LDK1]
    float* BW = S + 64 * LDK1;                                  // [64][LDK1]
    int*   PY = (int*)(BW + 64 * LDK1);                         // [64]
    int*   PX = PY + 64;
    float* FY = (float*)(PX + 64);
    float* FX = FY + 64;

    const int bid = blockIdx.x;          // 2048 blocks = B*H*2
    const int cw  = bid & 1;
    const int h   = (bid >> 1) & (HH - 1);
    const int b   = bid >> 8;
    const int w0  = cw * 64;
    const int tid = threadIdx.x;

    // Phase A: per-pixel bilinear params (k=1, pad=0 -> base = (h,w))
    if (tid < 64) {
        const int w = w0 + tid;
        const float oy = off1[(((size_t)b * 2 + 0) * HH + h) * WW + w];
        const float ox = off1[(((size_t)b * 2 + 1) * HH + h) * WW + w];
        const float py = oy + (float)h;
        const float px = ox + (float)w;
        const float fy = floorf(py), fx = floorf(px);
        PY[tid] = (int)fy; PX[tid] = (int)fx;
        FY[tid] = py - fy; FX[tid] = px - fx;
    }

    // Stage w1 into LDS (coalesced float4 global loads -> b128 LDS stores).
    // flat g = n*192 + k ; n = g4/48 ; dest (n*196 + k)*4 is 16B aligned.
    for (int g4 = tid; g4 < (CO * CIN) / 4; g4 += 128) {
        const int n = g4 / 48;
        const int k = 4 * g4 - n * CIN;
        *(float4*)&BW[n * LDK1 + k] = *(const float4*)&w1[4 * g4];
    }
    __syncthreads();

    // Phase B: gather + blend 64 px x 192 ch into LDS
    {
        const int p  = tid >> 1;
        const int c0 = (tid & 1) * 96;
        const int y0 = PY[p], x0 = PX[p];
        const float wy = FY[p], wx = FX[p];
        const bool y0v = (y0 >= 0) && (y0 < HH);
        const bool y1v = (y0 >= -1) && (y0 < HH - 1);
        const bool x0v = (x0 >= 0) && (x0 < WW);
        const bool x1v = (x0 >= -1) && (x0 < WW - 1);
        const int yc0 = min(max(y0, 0), HH - 1);
        const int yc1 = min(max(y0 + 1, 0), HH - 1);
        const int xc0 = min(max(x0, 0), WW - 1);
        const int xc1 = min(max(x0 + 1, 0), WW - 1);
        const float m00 = ((y0v && x0v) ? 1.f : 0.f) * (1.f - wy) * (1.f - wx);
        const float m01 = ((y0v && x1v) ? 1.f : 0.f) * (1.f - wy) * wx;
        const float m10 = ((y1v && x0v) ? 1.f : 0.f) * wy * (1.f - wx);
        const float m11 = ((y1v && x1v) ? 1.f : 0.f) * wy * wx;
        const int i00 = yc0 * WW + xc0, i01 = yc0 * WW + xc1;
        const int i10 = yc1 * WW + xc0, i11 = yc1 * WW + xc1;
        const size_t plane = (size_t)HH * WW;
        for (int cc = 0; cc < 96; ++cc) {
            const int c = c0 + cc;
            const float* pl =
                (c < 64)  ? (x1 + ((size_t)b * 64 + c) * plane)
              : (c < 128) ? (x2 + ((size_t)b * 64 + (c - 64)) * plane)
                          : (x4 + ((size_t)b * 64 + (c - 128)) * plane);
            S[p * LDK1 + c] = pl[i00] * m00 + pl[i01] * m01 +
                              pl[i10] * m10 + pl[i11] * m11;
        }
    }
    __syncthreads();

    // Phase C: WMMA. Wave wv owns pixels [16*wv, 16*wv+16), all N=64.
    const int lane  = tid & 31;
    const int wv    = tid >> 5;
    const int p0    = wv * 16;
    const int mrow  = lane & 15;
    const int khalf = (lane >> 4) << 1;    // A/B lanes 16-31 hold K+2,K+3

    v8f acc0 = {0,0,0,0,0,0,0,0};
    v8f acc1 = acc0, acc2 = acc0, acc3 = acc0;

    const float* Arow = &S[(p0 + mrow) * LDK1 + khalf];
    const float* B0   = &BW[(0 * 16 + mrow) * LDK1 + khalf];
    const float* B1   = &BW[(1 * 16 + mrow) * LDK1 + khalf];
    const float* B2   = &BW[(2 * 16 + mrow) * LDK1 + khalf];
    const float* B3   = &BW[(3 * 16 + mrow) * LDK1 + khalf];

    for (int k0 = 0; k0 < CIN; k0 += 4) {
        const v2f a   = *(const v2f*)(Arow + k0);
        const v2f bf0 = *(const v2f*)(B0 + k0);
        const v2f bf1 = *(const v2f*)(B1 + k0);
        const v2f bf2 = *(const v2f*)(B2 + k0);
        const v2f bf3 = *(const v2f*)(B3 + k0);
        acc0 = __builtin_amdgcn_wmma_f32_16x16x4_f32(false, a, false, bf0, (short)0, acc0, false, false);
        acc1 = __builtin_amdgcn_wmma_f32_16x16x4_f32(false, a, false, bf1, (short)0, acc1, false, false);
        acc2 = __builtin_amdgcn_wmma_f32_16x16x4_f32(false, a, false, bf2, (short)0, acc2, false, false);
        acc3 = __builtin_amdgcn_wmma_f32_16x16x4_f32(false, a, false, bf3, (short)0, acc3, false, false);
    }

    // Epilogue: fused conv-bias + BN + ReLU.  C/D layout: N=lane%16, M=r+8*(lane>>4).
    const int mofs = (lane >> 4) * 8;
    v8f accs[4] = { acc0, acc1, acc2, acc3 };
#pragma unroll
    for (int t = 0; t < 4; ++t) {
        const int n = t * 16 + mrow;
        const float inv  = bng[n] * rsqrtf(bnv[n] + 1e-5f);
        const float bias = b1[n] * inv + bnb[n] - bnm[n] * inv;
#pragma unroll
        for (int r = 0; r < 8; ++r) {
            float val = accs[t][r] * inv + bias;
            val = fmaxf(val, 0.f);
            y1[(((size_t)b * CO + n) * HH + h) * WW + (w0 + p0 + mofs + r)] = val;
        }
    }
}

// ---------------------------------------------------------------------------
// Kernel 3: off2 = 3x3 conv 64 -> 18, pad 1 (weights uniform -> scalar loads)
// ---------------------------------------------------------------------------
__global__ void __launch_bounds__(256)
k_off2(const float* __restrict__ yin, const float* __restrict__ wt,
       const float* __restrict__ bias, float* __restrict__ off2)
{
    const int idx = blockIdx.x * blockDim.x + threadIdx.x;
    const int w = idx & (WW - 1);
    const int h = (idx >> 7) & (HH - 1);
    const int b = idx >> 14;

    float acc[18];
#pragma unroll
    for (int o = 0; o < 18; ++o) acc[o] = 0.f;

    const size_t plane = (size_t)HH * WW;
    for (int c = 0; c < 64; ++c) {
        const float* pl = yin + ((size_t)b * CO + c) * plane;
#pragma unroll
        for (int ty = 0; ty < 3; ++ty) {
            const int yy = h - 1 + ty;
            if ((unsigned)yy >= HH) continue;
#pragma unroll
            for (int tx = 0; tx < 3; ++tx) {
                const int xx = w - 1 + tx;
                if ((unsigned)xx >= WW) continue;
                const float v = pl[yy * WW + xx];
#pragma unroll
                for (int o = 0; o < 18; ++o)
                    acc[o] = fmaf(v, wt[((o * 64 + c) * 3 + ty) * 3 + tx], acc[o]);
            }
        }
    }
#pragma unroll
    for (int o = 0; o < 18; ++o)
        off2[(((size_t)b * 18 + o) * HH + h) * WW + w] = acc[o] + bias[o];
}

// ---------------------------------------------------------------------------
// Kernel 4: deform-conv k=3 (K=576 -> 64 GEMM via WMMA f32 16x16x4) + BN
// Block: 128 threads (4 waves), tile = 32 pixels; waves split 2xM x 2xN.
// w2 staged in LDS in 4 K-chunks of 144; inner loop is pure DS + WMMA.
// ---------------------------------------------------------------------------
__global__ void __launch_bounds__(128)
k_deform2(const float* __restrict__ yin, const float* __restrict__ off2,
          const float* __restrict__ w2, const float* __restrict__ b2,
          const float* __restrict__ bng, const float* __restrict__ bnb,
          const float* __restrict__ bnm, const float* __restrict__ bnv,
          float* __restrict__ out)
{
    extern __shared__ char smem[];
    float* S  = (float*)smem;                                   // [32][LDK2]
    float* BW = S + 32 * LDK2;                                  // [64][LDB4]
    int*   PY = (int*)(BW + 64 * LDB4);                         // [32*9]
    int*   PX = PY + 288;
    float* FY = (float*)(PX + 288);
    float* FX = FY + 288;

    const int bid = blockIdx.x;          // 4096 blocks = B*H*4
    const int cw  = bid & 3;
    const int h   = (bid >> 2) & (HH - 1);
    const int b   = bid >> 9;
    const int w0  = cw * 32;
    const int tid = threadIdx.x;

    // Phase A: per-(pixel,tap) bilinear params (k=3, pad=1)
    for (int idx = tid; idx < 288; idx += 128) {
        const int p = idx / 9;
        const int t = idx - p * 9;
        const int ky = t / 3, kx = t - ky * 3;
        const int w = w0 + p;
        const float oy = off2[(((size_t)b * 18 + 2 * t    ) * HH + h) * WW + w];
        const float ox = off2[(((size_t)b * 18 + 2 * t + 1) * HH + h) * WW + w];
        const float py = oy + (float)(h - 1 + ky);
        const float px = ox + (float)(w - 1 + kx);
        const float fy = floorf(py), fx = floorf(px);
        PY[idx] = (int)fy; PX[idx] = (int)fx;
        FY[idx] = py - fy; FX[idx] = px - fx;
    }
    __syncthreads();

    // Phase B: fill S[p][c*9+t] = bilinear sample of channel c at tap t
    const size_t plane = (size_t)HH * WW;
    for (int p = 0; p < 32; ++p) {
        for (int idx = tid; idx < 576; idx += 128) {
            const int c = idx / 9;
            const int t = idx - c * 9;
            const int pi = p * 9 + t;
            const int y0 = PY[pi], x0 = PX[pi];
            const float wy = FY[pi], wx = FX[pi];
            const bool y0v = (y0 >= 0) && (y0 < HH);
            const bool y1v = (y0 >= -1) && (y0 < HH - 1);
            const bool x0v = (x0 >= 0) && (x0 < WW);
            const bool x1v = (x0 >= -1) && (x0 < WW - 1);
            const int yc0 = min(max(y0, 0), HH - 1);
            const int yc1 = min(max(y0 + 1, 0), HH - 1);
            const int xc0 = min(max(x0, 0), WW - 1);
            const int xc1 = min(max(x0 + 1, 0), WW - 1);
            const float m00 = ((y0v && x0v) ? 1.f : 0.f) * (1.f - wy) * (1.f - wx);
            const float m01 = ((y0v && x1v) ? 1.f : 0.f) * (1.f - wy) * wx;
            const float m10 = ((y1v && x0v) ? 1.f : 0.f) * wy * (1.f - wx);
            const float m11 = ((y1v && x1v) ? 1.f : 0.f) * wy * wx;
            const float* pl = yin + ((size_t)b * CO + c) * plane;
            S[p * LDK2 + idx] =
                pl[yc0 * WW + xc0] * m00 + pl[yc0 * WW + xc1] * m01 +
                pl[yc1 * WW + xc0] * m10 + pl[yc1 * WW + xc1] * m11;
        }
    }
    __syncthreads();

    // Phase C: WMMA over 4 K-chunks of 144, weights staged per chunk.
    const int lane  = tid & 31;
    const int wv    = tid >> 5;
    const int p0    = (wv & 1) * 16;
    const int nb    = (wv >> 1) * 32;
    const int mrow  = lane & 15;
    const int khalf = (lane >> 4) << 1;
    const int mofs  = (lane >> 4) * 8;

    v8f acc0 = {0,0,0,0,0,0,0,0};
    v8f acc1 = acc0;

    for (int kc = 0; kc < 4; ++kc) {
        // Stage chunk: BW[n][k] = w2[n*576 + kc*144 + k], k in [0,144)
        // flat g = n*144 + k ; n = g4/36 ; dest (n*148 + k)*4 is 16B aligned.
        for (int g4 = tid; g4 < (CO * 144) / 4; g4 += 128) {
            const int n = g4 / 36;
            const int k = 4 * g4 - n * 144;
            *(float4*)&BW[n * LDB4 + k] =
                *(const float4*)&w2[(size_t)n * 576 + kc * 144 + k];
        }
        __syncthreads();

        const float* Arow = &S[(p0 + mrow) * LDK2 + kc * 144 + khalf];
        const float* B0   = &BW[(nb + mrow     ) * LDB4 + khalf];
        const float* B1   = &BW[(nb + 16 + mrow) * LDB4 + khalf];
        for (int k0 = 0; k0 < 144; k0 += 4) {
            const v2f a   = *(const v2f*)(Arow + k0);
            const v2f bf0 = *(const v2f*)(B0 + k0);
            const v2f bf1 = *(const v2f*)(B1 + k0);
            acc0 = __builtin_amdgcn_wmma_f32_16x16x4_f32(false, a, false, bf0, (short)0, acc0, false, false);
            acc1 = __builtin_amdgcn_wmma_f32_16x16x4_f32(false, a, false, bf1, (short)0, acc1, false, false);
        }
        __syncthreads();   // protect BW before next chunk's staging
    }

    // Epilogue: fused conv-bias + BN (no ReLU)
    v8f accs[2] = { acc0, acc1 };
#pragma unroll
    for (int t = 0; t < 2; ++t) {
        const int n = nb + t * 16 + mrow;
        const float inv  = bng[n] * rsqrtf(bnv[n] + 1e-5f);
        const float bias = b2[n] * inv + bnb[n] - bnm[n] * inv;
#pragma unroll
        for (int r = 0; r < 8; ++r) {
            out[(((size_t)b * CO + n) * HH + h) * WW + (w0 + p0 + mofs + r)] =
                accs[t][r] * inv + bias;
        }
    }
}

// ---------------------------------------------------------------------------
extern "C" void kernel_launch(void* const* d_in, const int* in_sizes, int n_in,
                              void* d_out, int out_size, void* d_ws, size_t ws_size,
                              hipStream_t stream)
{
    const float* x1     = (const float*)d_in[0];
    const float* x2     = (const float*)d_in[1];
    const float* x4     = (const float*)d_in[2];
    const float* off1_w = (const float*)d_in[3];
    const float* off1_b = (const float*)d_in[4];
    const float* w1     = (const float*)d_in[5];
    const float* b1     = (const float*)d_in[6];
    const float* bn1_g  = (const float*)d_in[7];
    const float* bn1_b  = (const float*)d_in[8];
    const float* bn1_m  = (const float*)d_in[9];
    const float* bn1_v  = (const float*)d_in[10];
    const float* off2_w = (const float*)d_in[11];
    const float* off2_b = (const float*)d_in[12];
    const float* w2     = (const float*)d_in[13];
    const float* b2     = (const float*)d_in[14];
    const float* bn2_g  = (const float*)d_in[15];
    const float* bn2_b  = (const float*)d_in[16];
    const float* bn2_m  = (const float*)d_in[17];
    const float* bn2_v  = (const float*)d_in[18];
    float* out = (float*)d_out;

    const size_t OFF1_BYTES = (size_t)BB_ * 2 * HH * WW * sizeof(float);   // 1 MB
    const size_t Y1_BYTES   = (size_t)BB_ * CO * HH * WW * sizeof(float);  // 32 MB
    float* off1 = (float*)d_ws;
    float* y1   = (float*)((char*)d_ws + OFF1_BYTES);
    float* off2 = (float*)((char*)d_ws + OFF1_BYTES + Y1_BYTES);

    const int npix = BB_ * HH * WW;                       // 131072

    // Stage 1: offset conv 1x1
    k_off1<<<npix / 256, 256, 0, stream>>>(x1, x2, x4, off1_w, off1_b, off1);

    // Stage 2: deform-conv k=1 + BN + ReLU (WMMA)
    const size_t lds2 = (size_t)2 * 64 * LDK1 * sizeof(float) + 64 * 16;
    (void)hipFuncSetAttribute((const void*)k_deform1,
                              hipFuncAttributeMaxDynamicSharedMemorySize, (int)lds2);
    k_deform1<<<BB_ * HH * 2, 128, lds2, stream>>>(
        x1, x2, x4, off1, w1, b1, bn1_g, bn1_b, bn1_m, bn1_v, y1);

    // Stage 3: offset conv 3x3
    k_off2<<<npix / 256, 256, 0, stream>>>(y1, off2_w, off2_b, off2);

    // Stage 4: deform-conv k=3 + BN (WMMA)
    const size_t lds4 = (size_t)32 * LDK2 * sizeof(float)
                      + (size_t)64 * LDB4 * sizeof(float) + 288 * 16;
    (void)hipFuncSetAttribute((const void*)k_deform2,
                              hipFuncAttributeMaxDynamicSharedMemorySize, (int)lds4);
    k_deform2<<<BB_ * HH * 4, 128, lds4, stream>>>(
        y1, off2, w2, b2, bn2_g, bn2_b, bn2_m, bn2_v, out);
}